// DSGIAT_Classifier_15831249453410
// MI455X (gfx1250) — compile-verified
//
#include <hip/hip_runtime.h>
#include <hip/hip_bf16.h>

typedef float v2f __attribute__((ext_vector_type(2)));
typedef float v8f __attribute__((ext_vector_type(8)));

#define NNODES 50000
#define NEDGES 800000
#define NGRAPH 128
#define HC 256          // HEADS*HID
#define HEADS 4
#define HID 64

static inline int cdiv(int a, int b) { return (a + b - 1) / b; }

// ---------- utility ----------
__global__ void fill_u32_kernel(unsigned* __restrict__ p, unsigned v, int n) {
    int i = blockIdx.x * blockDim.x + threadIdx.x;
    if (i < n) p[i] = v;
}

// order-preserving float <-> uint encoding for atomic max on signed floats
__device__ __forceinline__ unsigned fenc(float f) {
    unsigned u = __float_as_uint(f);
    return (u & 0x80000000u) ? ~u : (u | 0x80000000u);
}
__device__ __forceinline__ float fdec(unsigned e) {
    unsigned u = (e & 0x80000000u) ? (e & 0x7FFFFFFFu) : ~e;
    return __uint_as_float(u);
}

// ---------- f32 WMMA GEMM: Out[M,Nc] = A[M,K] @ W[K,Nc] ----------
// one wave32 per 16x16 output tile, V_WMMA_F32_16X16X4_F32 over K in steps of 4
__global__ void wmma_gemm_f32(const float* __restrict__ A,
                              const float* __restrict__ W,
                              float* __restrict__ Out,
                              int M, int K, int Nc) {
    int gtid  = blockIdx.x * blockDim.x + threadIdx.x;
    int wave  = gtid >> 5;
    int lane  = threadIdx.x & 31;
    int tilesN = Nc >> 4;
    int tm = wave / tilesN;
    int tn = wave - tm * tilesN;
    if (tm * 16 >= M) return;             // wave-uniform guard (EXEC stays all-ones)
    int row0 = tm * 16, col0 = tn * 16;
    int r  = lane & 15;                    // row (A) / col (B,C)
    int hi = lane >> 4;                    // lane half select

    v8f acc = {0.f, 0.f, 0.f, 0.f, 0.f, 0.f, 0.f, 0.f};
    for (int kb = 0; kb < K; kb += 4) {
        v2f a, b;
        // A 16x4 f32: lanes 0-15 -> K={kb,kb+1}; lanes 16-31 -> K={kb+2,kb+3}
        a.x = A[(size_t)(row0 + r) * K + kb + hi * 2 + 0];
        a.y = A[(size_t)(row0 + r) * K + kb + hi * 2 + 1];
        // B 4x16 f32: mirrored striping, cols striped across lanes
        b.x = W[(size_t)(kb + hi * 2 + 0) * Nc + col0 + r];
        b.y = W[(size_t)(kb + hi * 2 + 1) * Nc + col0 + r];
        acc = __builtin_amdgcn_wmma_f32_16x16x4_f32(false, a, false, b,
                                                    (short)0, acc, false, false);
    }
    // C/D: VGPR i -> rows i (lanes 0-15) / i+8 (lanes 16-31), col = lane&15
    #pragma unroll
    for (int i = 0; i < 8; ++i)
        Out[(size_t)(row0 + i + hi * 8) * Nc + col0 + r] = acc[i];
}

// ---------- attention logits: es/ed [N,4] ----------
__global__ void attn_logits_kernel(const float* __restrict__ h,
                                   const float* __restrict__ a_s,
                                   const float* __restrict__ a_d,
                                   float* __restrict__ es, float* __restrict__ ed, int N) {
    int t = blockIdx.x * blockDim.x + threadIdx.x;
    if (t >= N * HEADS) return;
    int n = t >> 2, hd = t & 3;
    const float* hp = h + (size_t)n * HC + hd * HID;
    const float* as = a_s + hd * HID;
    const float* ad = a_d + hd * HID;
    float s0 = 0.f, s1 = 0.f;
    for (int c = 0; c < HID; ++c) { float v = hp[c]; s0 += v * as[c]; s1 += v * ad[c]; }
    es[t] = s0; ed[t] = s1;
}

// ---------- edge logits + atomic segment max ----------
__global__ void edge_logits_kernel(const float* __restrict__ es, const float* __restrict__ ed,
                                   const int* __restrict__ src, const int* __restrict__ dst,
                                   float* __restrict__ ee, unsigned* __restrict__ mx, int E) {
    int t = blockIdx.x * blockDim.x + threadIdx.x;
    if (t >= E * HEADS) return;
    int e = t >> 2, hd = t & 3;
    int s = src[e], d = dst[e];
    float v = es[s * HEADS + hd] + ed[d * HEADS + hd];
    v = (v > 0.f) ? v : 0.2f * v;                 // leaky_relu(.,0.2)
    ee[t] = v;
    atomicMax(&mx[d * HEADS + hd], fenc(v));
}

// ---------- exp(e - max) + atomic segment sum ----------
__global__ void edge_exp_kernel(float* __restrict__ ee, const unsigned* __restrict__ mx,
                                const int* __restrict__ dst, float* __restrict__ denom, int E) {
    int t = blockIdx.x * blockDim.x + threadIdx.x;
    if (t >= E * HEADS) return;
    int e = t >> 2, hd = t & 3;
    int d = dst[e];
    float w = __expf(ee[t] - fdec(mx[d * HEADS + hd]));
    ee[t] = w;
    atomicAdd(&denom[d * HEADS + hd], w);
}

// ---------- GAT aggregation: wave per edge, stride-32 channels ----------
__global__ void gat_aggregate_kernel(const float* __restrict__ h, const float* __restrict__ ee,
                                     const float* __restrict__ denom,
                                     const int* __restrict__ src, const int* __restrict__ dst,
                                     float* __restrict__ out, int E) {
    int gtid = blockIdx.x * blockDim.x + threadIdx.x;
    int e = gtid >> 5;
    if (e >= E) return;
    int lane = threadIdx.x & 31;
    int s = src[e], d = dst[e];
    float al[HEADS];
    #pragma unroll
    for (int hd = 0; hd < HEADS; ++hd) {
        float dn = denom[d * HEADS + hd];
        al[hd] = ee[e * HEADS + hd] / (dn > 0.f ? dn : 1.f);
    }
    const float* hs = h + (size_t)s * HC;
    float* od = out + (size_t)d * HC;
    #pragma unroll
    for (int i = 0; i < 8; ++i) {
        int c = lane + 32 * i;                    // coalesced 128B per wave
        atomicAdd(&od[c], hs[c] * al[i >> 1]);    // head = c>>6 = i>>1
    }
}

// ---------- out = relu(in + bias[c]) ----------
__global__ void bias_relu_kernel(const float* __restrict__ in, const float* __restrict__ b,
                                 float* __restrict__ out, int N) {
    int t = blockIdx.x * blockDim.x + threadIdx.x;
    if (t >= N * HC) return;
    int c = t & (HC - 1);
    float v = in[t] + b[c];
    out[t] = v > 0.f ? v : 0.f;
}

// ---------- label prop: degree / inv-sqrt / diffusion / combine ----------
__global__ void deg_kernel(const int* __restrict__ dst, float* __restrict__ deg, int E) {
    int t = blockIdx.x * blockDim.x + threadIdx.x;
    if (t < E) atomicAdd(&deg[dst[t]], 1.0f);
}
__global__ void dis_kernel(float* __restrict__ deg, int N) {
    int t = blockIdx.x * blockDim.x + threadIdx.x;
    if (t >= N) return;
    float d = deg[t];
    deg[t] = d > 0.f ? rsqrtf(fmaxf(d, 1.0f)) : 0.f;
}
__global__ void lp_aggregate_kernel(const float* __restrict__ xin, const float* __restrict__ dis,
                                    const int* __restrict__ src, const int* __restrict__ dst,
                                    float* __restrict__ T, int E) {
    int gtid = blockIdx.x * blockDim.x + threadIdx.x;
    int e = gtid >> 5;
    if (e >= E) return;
    int lane = threadIdx.x & 31;
    int s = src[e], d = dst[e];
    float nrm = dis[s] * dis[d];
    const float* xs = xin + (size_t)s * HC;
    float* td = T + (size_t)d * HC;
    #pragma unroll
    for (int i = 0; i < 8; ++i) {
        int c = lane + 32 * i;
        atomicAdd(&td[c], xs[c] * nrm);
    }
}
__global__ void lp_combine_kernel(const float* __restrict__ T, const float* __restrict__ resx,
                                  float* __restrict__ out, int N) {
    int t = blockIdx.x * blockDim.x + threadIdx.x;
    if (t >= N * HC) return;
    float v = 0.5f * T[t] + 0.5f * resx[t];       // alpha = 0.5
    out[t] = fminf(fmaxf(v, 0.f), 1.f);
}

// ---------- pooling ----------
__global__ void pool_count_kernel(const int* __restrict__ batch, float* __restrict__ cnt, int N) {
    int t = blockIdx.x * blockDim.x + threadIdx.x;
    if (t < N) atomicAdd(&cnt[batch[t]], 1.0f);
}
__global__ void pool_sum_kernel(const float* __restrict__ x, const float* __restrict__ h1,
                                const float* __restrict__ h2, const int* __restrict__ batch,
                                float* __restrict__ g, int N) {
    int t = blockIdx.x * blockDim.x + threadIdx.x;
    if (t >= N * 576) return;
    int n = t / 576, c = t - n * 576;
    float v;
    if (c < 64)       v = x [(size_t)n * 64  + c];
    else if (c < 320) v = h1[(size_t)n * HC + (c - 64)];
    else              v = h2[(size_t)n * HC + (c - 320)];
    atomicAdd(&g[batch[n] * 576 + c], v);
}
__global__ void pool_div_kernel(float* __restrict__ g, const float* __restrict__ cnt, int G) {
    int t = blockIdx.x * blockDim.x + threadIdx.x;
    if (t >= G * 576) return;
    g[t] /= fmaxf(cnt[t / 576], 1.0f);
}

// ---------- small dense layer: Out[M,Nn] = act(A[M,lda] @ W[K,Nn] + bias) ----------
__global__ void small_gemm_kernel(const float* __restrict__ A, int lda,
                                  const float* __restrict__ W, const float* __restrict__ bias,
                                  float* __restrict__ Out, int ldo,
                                  int M, int K, int Nn, int relu) {
    int t = blockIdx.x * blockDim.x + threadIdx.x;
    if (t >= M * Nn) return;
    int r = t / Nn, c = t - r * Nn;
    float acc = bias[c];
    for (int k = 0; k < K; ++k) acc += A[(size_t)r * lda + k] * W[(size_t)k * Nn + c];
    if (relu) acc = fmaxf(acc, 0.f);
    Out[(size_t)r * ldo + c] = acc;
}

// =====================================================================

extern "C" void kernel_launch(void* const* d_in, const int* in_sizes, int n_in,
                              void* d_out, int out_size, void* d_ws, size_t ws_size,
                              hipStream_t stream) {
    const int N = NNODES, E = NEDGES, G = NGRAPH;
    const float* x     = (const float*)d_in[0];
    const int*   ei    = (const int*)  d_in[1];
    const int*   batch = (const int*)  d_in[2];
    const float* clin  = (const float*)d_in[3];
    const float *W1 = (const float*)d_in[4],  *a1s = (const float*)d_in[5],
                *a1d = (const float*)d_in[6], *b1  = (const float*)d_in[7];
    const float *W2 = (const float*)d_in[8],  *a2s = (const float*)d_in[9],
                *a2d = (const float*)d_in[10],*b2  = (const float*)d_in[11];
    const float *pp_w1=(const float*)d_in[12],*pp_b1=(const float*)d_in[13];
    const float *pp_w2=(const float*)d_in[14],*pp_b2=(const float*)d_in[15];
    const float *cl_w1=(const float*)d_in[16],*cl_b1=(const float*)d_in[17];
    const float *cl_w2=(const float*)d_in[18],*cl_b2=(const float*)d_in[19];
    const float *h_w1=(const float*)d_in[20], *h_b1=(const float*)d_in[21];
    const float *h_w2=(const float*)d_in[22], *h_b2=(const float*)d_in[23];
    const float *h_w3=(const float*)d_in[24], *h_b3=(const float*)d_in[25];
    const int* srcp = ei;
    const int* dstp = ei + E;

    // workspace carve-out
    size_t off = 0;
    char* base = (char*)d_ws;
    auto carve = [&](size_t bytes) -> char* {
        char* p = base + off;
        off += (bytes + 255) & ~(size_t)255;
        return p;
    };
    float*    hbuf  = (float*)   carve((size_t)N * HC * 4);  // x@W / post-bias-relu features
    float*    aggT  = (float*)   carve((size_t)N * HC * 4);  // scatter accumulator
    float*    lpB   = (float*)   carve((size_t)N * HC * 4);  // label-prop intermediate
    float*    h1    = (float*)   carve((size_t)N * HC * 4);  // layer-1 final (persists)
    float*    h2    = (float*)   carve((size_t)N * HC * 4);  // layer-2 final (persists)
    float*    es    = (float*)   carve((size_t)N * HEADS * 4);
    float*    ed    = (float*)   carve((size_t)N * HEADS * 4);
    unsigned* mx    = (unsigned*)carve((size_t)N * HEADS * 4);
    float*    denom = (float*)   carve((size_t)N * HEADS * 4);
    float*    eedge = (float*)   carve((size_t)E * HEADS * 4);
    float*    dis   = (float*)   carve((size_t)N * 4);
    float*    g     = (float*)   carve((size_t)G * 576 * 4);
    float*    cnt   = (float*)   carve((size_t)G * 4);
    float*    t256  = (float*)   carve((size_t)G * 256 * 4);
    float*    z     = (float*)   carve((size_t)G * 160 * 4);
    float*    t64   = (float*)   carve((size_t)G * 64 * 4);
    float*    t32   = (float*)   carve((size_t)G * 32 * 4);

    const int TB = 256;
    const unsigned ENC_NEG_INF = 0x007FFFFFu;  // fenc(-inf)
    auto fill = [&](void* p, unsigned v, int n) {
        fill_u32_kernel<<<cdiv(n, TB), TB, 0, stream>>>((unsigned*)p, v, n);
    };

    // shared sym-norm degree (same graph for both label_prop calls)
    fill(dis, 0u, N);
    deg_kernel<<<cdiv(E, TB), TB, 0, stream>>>(dstp, dis, E);
    dis_kernel<<<cdiv(N, TB), TB, 0, stream>>>(dis, N);

    auto gat_plus_lp = [&](const float* xin, int K, const float* W,
                           const float* as_, const float* ad_, const float* bias,
                           float* hfinal) {
        // h = xin @ W   (WMMA f32)
        int waves = (N / 16) * (HC / 16);
        wmma_gemm_f32<<<cdiv(waves * 32, TB), TB, 0, stream>>>(xin, W, hbuf, N, K, HC);
        // attention
        attn_logits_kernel<<<cdiv(N * HEADS, TB), TB, 0, stream>>>(hbuf, as_, ad_, es, ed, N);
        fill(mx, ENC_NEG_INF, N * HEADS);
        fill(denom, 0u, N * HEADS);
        edge_logits_kernel<<<cdiv(E * HEADS, TB), TB, 0, stream>>>(es, ed, srcp, dstp, eedge, mx, E);
        edge_exp_kernel<<<cdiv(E * HEADS, TB), TB, 0, stream>>>(eedge, mx, dstp, denom, E);
        fill(aggT, 0u, N * HC);
        gat_aggregate_kernel<<<cdiv(E * 32, TB), TB, 0, stream>>>(hbuf, eedge, denom, srcp, dstp, aggT, E);
        bias_relu_kernel<<<cdiv(N * HC, TB), TB, 0, stream>>>(aggT, bias, hbuf, N);  // hbuf = relu(gat)+... input to LP
        // label prop: 2 iterations, residual vs hbuf
        fill(aggT, 0u, N * HC);
        lp_aggregate_kernel<<<cdiv(E * 32, TB), TB, 0, stream>>>(hbuf, dis, srcp, dstp, aggT, E);
        lp_combine_kernel<<<cdiv(N * HC, TB), TB, 0, stream>>>(aggT, hbuf, lpB, N);
        fill(aggT, 0u, N * HC);
        lp_aggregate_kernel<<<cdiv(E * 32, TB), TB, 0, stream>>>(lpB, dis, srcp, dstp, aggT, E);
        lp_combine_kernel<<<cdiv(N * HC, TB), TB, 0, stream>>>(aggT, hbuf, hfinal, N);
    };

    gat_plus_lp(x,  64,  W1, a1s, a1d, b1, h1);
    gat_plus_lp(h1, 256, W2, a2s, a2d, b2, h2);

    // mean pool of concat([x, h1, h2]) -> g [G,576]
    fill(g, 0u, G * 576);
    fill(cnt, 0u, G);
    pool_count_kernel<<<cdiv(N, TB), TB, 0, stream>>>(batch, cnt, N);
    pool_sum_kernel<<<cdiv(N * 576, TB), TB, 0, stream>>>(x, h1, h2, batch, g, N);
    pool_div_kernel<<<cdiv(G * 576, TB), TB, 0, stream>>>(g, cnt, G);

    // MLP head
    small_gemm_kernel<<<cdiv(G * 256, TB), TB, 0, stream>>>(g, 576, pp_w1, pp_b1, t256, 256, G, 576, 256, 1);
    small_gemm_kernel<<<cdiv(G * 128, TB), TB, 0, stream>>>(t256, 256, pp_w2, pp_b2, z, 160, G, 256, 128, 0);
    small_gemm_kernel<<<cdiv(G * 64, TB), TB, 0, stream>>>(clin, 32, cl_w1, cl_b1, t64, 64, G, 32, 64, 1);
    small_gemm_kernel<<<cdiv(G * 32, TB), TB, 0, stream>>>(t64, 64, cl_w2, cl_b2, z + 128, 160, G, 64, 32, 0);
    small_gemm_kernel<<<cdiv(G * 64, TB), TB, 0, stream>>>(z, 160, h_w1, h_b1, t64, 64, G, 160, 64, 1);
    small_gemm_kernel<<<cdiv(G * 32, TB), TB, 0, stream>>>(t64, 64, h_w2, h_b2, t32, 32, G, 64, 32, 1);
    small_gemm_kernel<<<cdiv(G * 2, TB), TB, 0, stream>>>(t32, 32, h_w3, h_b3, (float*)d_out, 2, G, 32, 2, 0);

    (void)in_sizes; (void)n_in; (void)out_size; (void)ws_size;
}